// CPAB_transformer_81080392613942
// MI455X (gfx1250) — compile-verified
//
#include <hip/hip_runtime.h>
#include <hip/hip_bf16.h>
#include <math.h>

// CPAB 1D transformer: NC=32, NSTEPS=32, n_theta=8, d=30, n_points=262144.
//
// Kernel 1 (1 wave): Avees[64][8] = basis[64][30] @ theta^T[30][8] via
//   V_WMMA_F32_16X16X4_F32 (A/B zero-padded in LDS -> unconditional ds reads),
//   then closed-form expm -> 256-entry (c00, NC*c01) table, all in one pass.
// Kernel 2: 2M trajectories x 32 steps; table in LDS; state kept as y = NC*x
//   (bit-exact power-of-two rescale) so each step is cvt/clamp/ds_load_b64/fma.

typedef __attribute__((ext_vector_type(2))) float v2f;
typedef __attribute__((ext_vector_type(8))) float v8f;

#define NC_      32
#define NSTEPS_  32
#define NTHETA_  8
#define DDIM_    30
#define NROWS_   64   // (NDIM+1)*NDIM*NC = 2*32
#define KPAD_    32   // K padded 30 -> 32
#define NPAD_    16   // N padded 8 -> 16

// ---------------------------------------------------------------------------
// Setup: WMMA GEMM + expm table, single wave32 workgroup.
// A (16x4 f32): lanes 0-15 rows M, VGPR0/1 = K0/K1; lanes 16-31 = K2/K3.
// B (4x16 f32): lanes 0-15 N=lane, VGPR0/1 = K0/K1; lanes 16-31 = K2/K3.
// D (16x16 f32): VGPR j -> M=j (lanes 0-15) / M=j+8 (lanes 16-31), N=lane&15.
// ---------------------------------------------------------------------------
__global__ __launch_bounds__(32)
void cpab_setup_kernel(const float* __restrict__ basis,   // [64][30]
                       const float* __restrict__ theta,   // [8][30]
                       float2* __restrict__ table) {      // [256] = [t][c]
    __shared__ float sB [NROWS_ * KPAD_];   // basis, zero-padded K   (8 KB)
    __shared__ float sTh[NPAD_  * KPAD_];   // theta, zero-padded N,K (2 KB)
    __shared__ float sAv[NROWS_ * NPAD_];   // Avees, padded N        (4 KB)

    const int lane = threadIdx.x;           // 0..31

    // Stage inputs into LDS with zero padding (unconditional loads + select).
    for (int idx = lane; idx < NROWS_ * KPAD_; idx += 32) {
        const int r = idx >> 5, k = idx & 31;
        const float v = basis[r * DDIM_ + (k < DDIM_ ? k : 0)];
        sB[idx] = (k < DDIM_) ? v : 0.0f;
    }
    for (int idx = lane; idx < NPAD_ * KPAD_; idx += 32) {
        const int n = idx >> 5, k = idx & 31;
        const bool valid = (n < NTHETA_) && (k < DDIM_);
        const float v = theta[valid ? (n * DDIM_ + k) : 0];
        sTh[idx] = valid ? v : 0.0f;
    }
    __syncthreads();

    const int half = lane >> 4;
    const int l    = lane & 15;

    for (int mt = 0; mt < 4; ++mt) {
        v8f acc = {};
        const int row = mt * 16 + l;
#pragma unroll
        for (int kk = 0; kk < 8; ++kk) {           // K = 32, 4 per WMMA
            const int k0 = kk * 4 + half * 2;
            v2f a, b;
            a.x = sB [row * KPAD_ + k0];
            a.y = sB [row * KPAD_ + k0 + 1];
            b.x = sTh[l   * KPAD_ + k0];
            b.y = sTh[l   * KPAD_ + k0 + 1];
            acc = __builtin_amdgcn_wmma_f32_16x16x4_f32(
                false, a, false, b, (short)0, acc, false, false);
        }
#pragma unroll
        for (int j = 0; j < 8; ++j) {
            const int m = j + 8 * half;            // D layout
            sAv[(mt * 16 + m) * NPAD_ + l] = acc[j];
        }
    }
    __syncthreads();

    // expm([[a,b],[0,0]]) -> (e^a, b*phi(a)), phi with reference's stable branch.
    // Store c01 pre-scaled by NC (exact power-of-two scale).
    const float dT = 1.0f / (float)NSTEPS_;
#pragma unroll
    for (int t = 0; t < NTHETA_; ++t) {
        const int c = lane;                        // cell 0..31
        const float a = dT * sAv[(2 * c    ) * NPAD_ + t];
        const float b = dT * sAv[(2 * c + 1) * NPAD_ + t];
        const float c00 = expf(a);
        const float phi = (fabsf(a) < 1e-6f) ? fmaf(0.5f, a, 1.0f)
                                             : (expm1f(a) / a);
        table[t * NC_ + c] = make_float2(c00, (float)NC_ * (b * phi));
    }
}

// ---------------------------------------------------------------------------
// Integrate: one point per thread, 8 independent theta-chains for ILP.
// State y = NC*x  =>  cell = clamp((int)y), y = fma(c00, y, NC*c01).
// Bit-exact vs reference (power-of-two rescale commutes with RN rounding).
// ---------------------------------------------------------------------------
__global__ __launch_bounds__(256)
void cpab_integrate_kernel(const float* __restrict__ points,
                           const float2* __restrict__ table,
                           float* __restrict__ out,
                           int n_points) {
    __shared__ float2 sT[NTHETA_ * NC_];   // 2 KB
    const int tid = threadIdx.x;
    sT[tid] = table[tid];
    __syncthreads();

    const int i = blockIdx.x * 256 + tid;
    if (i >= n_points) return;

    const float y0 = points[i] * (float)NC_;
    float y[NTHETA_];
#pragma unroll
    for (int t = 0; t < NTHETA_; ++t) y[t] = y0;

    for (int s = 0; s < NSTEPS_; ++s) {
#pragma unroll
        for (int t = 0; t < NTHETA_; ++t) {
            int cell = (int)y[t];                          // trunc==floor after clamp
            cell = cell < 0 ? 0 : (cell > NC_ - 1 ? NC_ - 1 : cell);
            const float2 T = sT[t * NC_ + cell];
            y[t] = fmaf(T.x, y[t], T.y);
        }
    }

#pragma unroll
    for (int t = 0; t < NTHETA_; ++t)
        out[t * n_points + i] = y[t] * (1.0f / (float)NC_);
}

// ---------------------------------------------------------------------------
extern "C" void kernel_launch(void* const* d_in, const int* in_sizes, int n_in,
                              void* d_out, int out_size, void* d_ws, size_t ws_size,
                              hipStream_t stream) {
    const float* points = (const float*)d_in[0];   // [1][262144]
    const float* theta  = (const float*)d_in[1];   // [8][30]
    const float* basis  = (const float*)d_in[2];   // [64][30]
    float* out = (float*)d_out;                    // [8][1][n_points] flat

    const int n_points = in_sizes[0];              // NDIM==1 -> flat count

    float2* table = (float2*)d_ws;                 // 256 * 8 B = 2 KB scratch

    cpab_setup_kernel<<<1, 32, 0, stream>>>(basis, theta, table);

    const int nblk = (n_points + 255) / 256;
    cpab_integrate_kernel<<<nblk, 256, 0, stream>>>(points, table, out, n_points);
}